// NPAttention_83811991814547
// MI455X (gfx1250) — compile-verified
//
#include <hip/hip_runtime.h>
#include <math.h>

// ---------------------------------------------------------------------------
// NPAttention fused pipeline for gfx1250 (MI455X).
// f32 WMMA 16x16x4 GEMMs + async global->LDS staging (ASYNCcnt) in the main
// GEMM, double-buffered K=16 stages.
// ---------------------------------------------------------------------------

typedef __attribute__((ext_vector_type(2))) float v2f;
typedef __attribute__((ext_vector_type(8))) float v8f;

#define HW   4096
#define CIN  256

static __device__ __forceinline__ v8f wmma_f32(v2f a, v2f b, v8f c) {
    // D = A(16x4) x B(4x16) + C(16x16), all f32, exact per reference math.
    return __builtin_amdgcn_wmma_f32_16x16x4_f32(false, a, false, b,
                                                 (short)0, c, false, false);
}

// Async 16B global->LDS copy (CDNA5, tracked by ASYNCcnt). lds_byte is the
// raw LDS byte offset (= low 32 bits of the flat shared-memory address).
static __device__ __forceinline__ void async_copy_b128(unsigned lds_byte,
                                                       const float* gaddr) {
    asm volatile("global_load_async_to_lds_b128 %0, %1, off"
                 :: "v"(lds_byte), "v"(gaddr)
                 : "memory");
}
static __device__ __forceinline__ void wait_async0() {
    asm volatile("s_wait_asynccnt 0x0" ::: "memory");
}

// --------------------------- K0: zero accumulators -------------------------
__global__ void np_init_zero(float* att, float* gnsum, float* gnsq) {
    int t = threadIdx.x;
    if (t < 128)       att[t]        = 0.0f;
    else if (t < 256)  gnsum[t-128]  = 0.0f;
    else if (t < 384)  gnsq[t-256]   = 0.0f;
}

// --------------- K1: fused GEMM  t = Wt*x (mask-modulated), p, g -----------
// Virtual weight matrix: rows [0,256)=Wt, [256,384)=Wp, [384,512)=Wg.
// grid = (HW/256, 512/32, B); block = 256 (8 waves: 2 M-subtiles x 4 N-subtiles)
// X tile staged through LDS with double-buffered async copies, K=16 per stage.
__global__ void np_fused_gemm_tpg(const float* __restrict__ x,
                                  const float* __restrict__ mask,
                                  const float* __restrict__ Wt,
                                  const float* __restrict__ Wtm,
                                  const float* __restrict__ Wp,
                                  const float* __restrict__ Wg,
                                  float* __restrict__ t_out,
                                  float* __restrict__ p_out,
                                  float* __restrict__ g_out) {
    __shared__ __align__(16) float xs[2][16][256];   // 32 KB double buffer

    const int b      = blockIdx.z;
    const int mBlock = blockIdx.y << 5;           // 0..480 step 32
    const int nBlock = blockIdx.x << 8;           // 0..3840 step 256
    const int tid    = threadIdx.x;
    const int wave   = tid >> 5;
    const int lane   = tid & 31;
    const int lrow   = lane & 15;
    const int khalf  = (lane >> 4) << 1;          // 0 or 2
    const int mWave  = mBlock + ((wave >> 2) << 4);
    const int nLoc   = (wave & 3) << 6;           // 0,64,128,192

    const float* W; int mLocal;
    if (mWave < 256)      { W = Wt; mLocal = mWave;       }
    else if (mWave < 384) { W = Wp; mLocal = mWave - 256; }
    else                  { W = Wg; mLocal = mWave - 384; }
    const float* wrow = W + (mLocal + lrow) * CIN;
    const float* xb   = x + (size_t)b * CIN * HW;

    // Async staging layout: each thread copies 4x B128 (16 floats) per stage.
    const unsigned ldsBase = (unsigned)(size_t)(&xs[0][0][0]);
    const int stq = tid >> 6;                     // 0..3  (row quarter)
    const int stn = (tid & 63) << 2;              // 0..252 step 4 (col base)

    v8f c0 = {}; v8f c1 = {}; v8f c2 = {}; v8f c3 = {};

    // Prologue: issue stage 0 into buffer 0.
    {
        const unsigned bo = ldsBase;
        #pragma unroll
        for (int q = 0; q < 4; ++q) {
            const int row = (q << 2) + stq;       // 0..15
            async_copy_b128(bo + (unsigned)(((row << 8) + stn) << 2),
                            xb + (size_t)row * HW + nBlock + stn);
        }
    }

    for (int ks = 0; ks < CIN; ks += 16) {
        const int buf = (ks >> 4) & 1;
        wait_async0();          // my copies into xs[buf] have landed
        __syncthreads();        // everyone's copies visible; prev reads done

        if (ks + 16 < CIN) {    // issue next stage into the other buffer
            const unsigned bo = ldsBase + (unsigned)((buf ^ 1) * 16 * 256 * 4);
            #pragma unroll
            for (int q = 0; q < 4; ++q) {
                const int row = (q << 2) + stq;
                async_copy_b128(bo + (unsigned)(((row << 8) + stn) << 2),
                                xb + (size_t)(ks + 16 + row) * HW + nBlock + stn);
            }
        }

        #pragma unroll
        for (int kk = 0; kk < 16; kk += 4) {
            const int k = ks + kk;
            v2f a;  a.x = wrow[k + khalf]; a.y = wrow[k + khalf + 1];
            const float* x0 = &xs[buf][kk + khalf    ][nLoc + lrow];
            const float* x1 = &xs[buf][kk + khalf + 1][nLoc + lrow];
            v2f b0; b0.x = x0[0];  b0.y = x1[0];
            v2f b1; b1.x = x0[16]; b1.y = x1[16];
            v2f b2; b2.x = x0[32]; b2.y = x1[32];
            v2f b3; b3.x = x0[48]; b3.y = x1[48];

            c0 = wmma_f32(a, b0, c0);
            c1 = wmma_f32(a, b1, c1);
            c2 = wmma_f32(a, b2, c2);
            c3 = wmma_f32(a, b3, c3);
        }
    }

    // Epilogue. C layout: vgpr r, lanes 0-15 -> M=r, lanes 16-31 -> M=8+r.
    const int mOff  = (lane >> 4) << 3;           // 0 or 8
    const int nl    = lane & 15;
    const int nBase = nBlock + nLoc + nl;

    if (mWave < 256) {
        const float wtm0 = Wtm[0], wtm1 = Wtm[1];
        const size_t mb = (size_t)b * 2 * HW;
        #pragma unroll
        for (int j = 0; j < 4; ++j) {
            const int n = nBase + (j << 4);
            const float s = 1.0f + wtm0 * mask[mb + n] + wtm1 * mask[mb + HW + n];
            v8f c = (j == 0) ? c0 : (j == 1) ? c1 : (j == 2) ? c2 : c3;
            #pragma unroll
            for (int r = 0; r < 8; ++r) {
                const int m = mWave + mOff + r;
                t_out[((size_t)b * 256 + m) * HW + n] = c[r] * s;
            }
        }
    } else {
        float* dst   = (mWave < 384) ? p_out : g_out;
        const int mL = (mWave < 384) ? (mWave - 256) : (mWave - 384);
        #pragma unroll
        for (int j = 0; j < 4; ++j) {
            const int n = nBase + (j << 4);
            v8f c = (j == 0) ? c0 : (j == 1) ? c1 : (j == 2) ? c2 : c3;
            #pragma unroll
            for (int r = 0; r < 8; ++r) {
                const int m = mL + mOff + r;
                dst[((size_t)b * 128 + m) * HW + n] = c[r];
            }
        }
    }
}

// ------------- K2: att[b,g] reduction with fused 3x3 maxpool ---------------
// grid = (HW/256, 128, B). Raw product -> att[b, c>>5]; pooled -> att[b, 4+(c>>5)].
__global__ void np_att_reduce(const float* __restrict__ p,
                              const float* __restrict__ g,
                              float* __restrict__ att) {
    __shared__ float sr[256];
    __shared__ float sq[256];
    const int b = blockIdx.z, c = blockIdx.y;
    const int n = (blockIdx.x << 8) + threadIdx.x;
    const int h = n >> 6, w = n & 63;
    const float* pr = p + ((size_t)b * 128 + c) * HW;
    const float* gr = g + ((size_t)b * 128 + c) * HW;

    const float raw = pr[n] * gr[n];
    float pm = -INFINITY, gm = -INFINITY;
    #pragma unroll
    for (int dh = -1; dh <= 1; ++dh) {
        const int hh = h + dh;
        if (hh < 0 || hh > 63) continue;
        #pragma unroll
        for (int dw = -1; dw <= 1; ++dw) {
            const int ww = w + dw;
            if (ww < 0 || ww > 63) continue;
            const int nn = (hh << 6) + ww;
            pm = fmaxf(pm, pr[nn]);
            gm = fmaxf(gm, gr[nn]);
        }
    }
    sr[threadIdx.x] = raw;
    sq[threadIdx.x] = pm * gm;
    __syncthreads();
    for (int s = 128; s > 0; s >>= 1) {
        if (threadIdx.x < s) {
            sr[threadIdx.x] += sr[threadIdx.x + s];
            sq[threadIdx.x] += sq[threadIdx.x + s];
        }
        __syncthreads();
    }
    if (threadIdx.x == 0) {
        atomicAdd(&att[b * 8 + (c >> 5)],     sr[0]);
        atomicAdd(&att[b * 8 + 4 + (c >> 5)], sq[0]);
    }
}

// ----------------- K3: per-row softmax statistics (max, 1/sum) -------------
// grid = B*256 rows
__global__ void np_softmax_stats(const float* __restrict__ t,
                                 float* __restrict__ rowmax,
                                 float* __restrict__ rowinv) {
    __shared__ float red[256];
    const int row = blockIdx.x;
    const float* tr = t + (size_t)row * HW;

    float m = -INFINITY;
    for (int i = threadIdx.x; i < HW; i += 256) m = fmaxf(m, tr[i]);
    red[threadIdx.x] = m; __syncthreads();
    for (int s = 128; s > 0; s >>= 1) {
        if (threadIdx.x < s)
            red[threadIdx.x] = fmaxf(red[threadIdx.x], red[threadIdx.x + s]);
        __syncthreads();
    }
    const float M = red[0];
    __syncthreads();

    float sum = 0.0f;
    for (int i = threadIdx.x; i < HW; i += 256) sum += __expf(tr[i] - M);
    red[threadIdx.x] = sum; __syncthreads();
    for (int s = 128; s > 0; s >>= 1) {
        if (threadIdx.x < s) red[threadIdx.x] += red[threadIdx.x + s];
        __syncthreads();
    }
    if (threadIdx.x == 0) { rowmax[row] = M; rowinv[row] = 1.0f / red[0]; }
}

// ----- K4: xz = att[b,g] * Wz[g] (32x32) x softmax(t) (32xHW) + GN stats ---
// grid = (HW/128, G, B); block = 256 (8 waves, each M=32 x N=16)
__global__ void np_xz_gemm_gn(const float* __restrict__ t,
                              const float* __restrict__ Wz,
                              const float* __restrict__ att,
                              const float* __restrict__ rowmax,
                              const float* __restrict__ rowinv,
                              float* __restrict__ xz,
                              float* __restrict__ gnsum,
                              float* __restrict__ gnsq) {
    __shared__ float s1[256];
    __shared__ float s2[256];
    const int b = blockIdx.z, g = blockIdx.y;
    const int lane  = threadIdx.x & 31;
    const int lrow  = lane & 15;
    const int khalf = (lane >> 4) << 1;
    const int nWave = (blockIdx.x << 7) + ((threadIdx.x >> 5) << 4);

    const float* Wzg  = Wz + g * 1024;                         // [o][i] row-major
    const float* trow = t + ((size_t)b * 256 + g * 32) * HW;
    const float* rm   = rowmax + b * 256 + g * 32;
    const float* ri   = rowinv + b * 256 + g * 32;
    const int n = nWave + lrow;

    v8f c0 = {}; v8f c1 = {};
    for (int k = 0; k < 32; k += 4) {
        const int i0 = k + khalf, i1 = i0 + 1;
        v2f bf;
        bf.x = __expf(trow[(size_t)i0 * HW + n] - rm[i0]) * ri[i0];
        bf.y = __expf(trow[(size_t)i1 * HW + n] - rm[i1]) * ri[i1];
        v2f a0; a0.x = Wzg[lrow * 32 + i0];        a0.y = Wzg[lrow * 32 + i1];
        v2f a1; a1.x = Wzg[(16 + lrow) * 32 + i0]; a1.y = Wzg[(16 + lrow) * 32 + i1];
        c0 = wmma_f32(a0, bf, c0);                 // output rows 0..15
        c1 = wmma_f32(a1, bf, c1);                 // output rows 16..31
    }

    const float av  = att[b * 8 + g];
    const int mOff  = (lane >> 4) << 3;
    const int no    = nWave + (lane & 15);
    float ls = 0.0f, lq = 0.0f;
    float* xrow = xz + ((size_t)b * 256 + g * 32) * HW;
    #pragma unroll
    for (int r = 0; r < 8; ++r) {
        const float v0 = c0[r] * av;
        const float v1 = c1[r] * av;
        xrow[(size_t)(mOff + r) * HW + no]      = v0;
        xrow[(size_t)(16 + mOff + r) * HW + no] = v1;
        ls += v0 + v1;
        lq += v0 * v0 + v1 * v1;
    }
    s1[threadIdx.x] = ls; s2[threadIdx.x] = lq;
    __syncthreads();
    for (int s = 128; s > 0; s >>= 1) {
        if (threadIdx.x < s) {
            s1[threadIdx.x] += s1[threadIdx.x + s];
            s2[threadIdx.x] += s2[threadIdx.x + s];
        }
        __syncthreads();
    }
    if (threadIdx.x == 0) {
        atomicAdd(&gnsum[b * 8 + g], s1[0]);
        atomicAdd(&gnsq[b * 8 + g],  s2[0]);
    }
}

// ------------------ K5: GroupNorm apply + affine + residual ----------------
__global__ void np_finalize(const float* __restrict__ xz,
                            const float* __restrict__ x,
                            const float* __restrict__ gnsum,
                            const float* __restrict__ gnsq,
                            const float* __restrict__ gamma,
                            const float* __restrict__ beta,
                            float* __restrict__ out) {
    const size_t idx = (size_t)blockIdx.x * 256 + threadIdx.x;
    const int b = (int)(idx >> 20);            // 256*4096 = 2^20 per batch
    const int c = (int)((idx >> 12) & 255);
    const int g = c >> 5;
    const float cnt = 131072.0f;               // 32 * 4096
    const float mu  = gnsum[b * 8 + g] / cnt;
    const float var = gnsq[b * 8 + g] / cnt - mu * mu;
    const float inv = rsqrtf(var + 1e-5f);
    out[idx] = (xz[idx] - mu) * inv * gamma[c] + beta[c] + x[idx];
}

// ---------------------------------------------------------------------------
extern "C" void kernel_launch(void* const* d_in, const int* in_sizes, int n_in,
                              void* d_out, int out_size, void* d_ws, size_t ws_size,
                              hipStream_t stream) {
    const float* x     = (const float*)d_in[0];
    const float* mask  = (const float*)d_in[1];
    const float* Wt    = (const float*)d_in[2];
    const float* Wtm   = (const float*)d_in[3];
    const float* Wp    = (const float*)d_in[4];
    const float* Wg    = (const float*)d_in[5];
    const float* Wz    = (const float*)d_in[6];
    const float* gamma = (const float*)d_in[7];
    const float* beta  = (const float*)d_in[8];
    float* out = (float*)d_out;

    const size_t NT = (size_t)16 * 256 * 4096;   // 16,777,216 elems
    float* ws     = (float*)d_ws;
    float* t_mod  = ws;                           // [B,256,HW]
    float* p_raw  = ws + NT;                      // [B,128,HW]
    float* g_raw  = ws + NT + NT / 2;             // [B,128,HW]
    float* xz     = ws + NT;                      // aliases p/g (used after att)
    float* attb   = ws + 2 * NT;                  // 128
    float* rowmax = attb + 128;                   // 4096
    float* rowinv = rowmax + 4096;                // 4096
    float* gnsum  = rowinv + 4096;                // 128
    float* gnsq   = gnsum + 128;                  // 128

    np_init_zero<<<1, 384, 0, stream>>>(attb, gnsum, gnsq);
    np_fused_gemm_tpg<<<dim3(16, 16, 16), 256, 0, stream>>>(
        x, mask, Wt, Wtm, Wp, Wg, t_mod, p_raw, g_raw);
    np_att_reduce<<<dim3(16, 128, 16), 256, 0, stream>>>(p_raw, g_raw, attb);
    np_softmax_stats<<<4096, 256, 0, stream>>>(t_mod, rowmax, rowinv);
    np_xz_gemm_gn<<<dim3(32, 8, 16), 256, 0, stream>>>(
        t_mod, Wz, attb, rowmax, rowinv, xz, gnsum, gnsq);
    np_finalize<<<65536, 256, 0, stream>>>(xz, x, gnsum, gnsq, gamma, beta, out);
}